// FullSparse_31748398252182
// MI455X (gfx1250) — compile-verified
//
#include <hip/hip_runtime.h>

// Sparse embedding-bag: out[r,:] = relu( sum_{k in seg(r)} values[k]*weight[col_ids[k],:] + bias )
// BATCH=4096, VOCAB=1e6, OUT=128, NNZ=409600, f32. Memory-bound (~0.48 FLOP/B; 210MB of
// random 512B row gathers dominate). CDNA5 path: async global->LDS DMA, double-buffered.

#define OUT_DIM 128

constexpr int WAVES_PER_BLOCK = 4;   // 128 threads/block
constexpr int CHUNK = 8;             // async rows per buffer (x2 buffers = 16 in flight/wave)

// ---------------------------------------------------------------------------
// Kernel A: build row_start[0..batch] (CSR offsets) from sorted row_ids in one
// parallel pass — no dependent-load chains, ~3MB traffic. row_start[r] is the
// first k with row_ids[k] >= r. Every entry is written exactly once per call.
// ---------------------------------------------------------------------------
__global__ void build_offsets_kernel(const int* __restrict__ row_ids,
                                     int* __restrict__ row_start,
                                     int nnz, int batch)
{
    const int k = blockIdx.x * blockDim.x + threadIdx.x;
    if (k >= nnz) return;
    const int r0 = row_ids[k];
    const int r1 = (k + 1 < nnz) ? row_ids[k + 1] : batch;
    if (k == 0) {
        for (int r = 0; r <= r0; ++r) row_start[r] = 0;
    }
    for (int r = r0 + 1; r <= r1; ++r) row_start[r] = k + 1;   // usually 0 or 1 iteration
}

// ---------------------------------------------------------------------------
// Kernel B: one wave per output row; lane owns 4 columns. Each async instruction
// DMAs one full 512B weight row (32 lanes x b128) into LDS.
// ---------------------------------------------------------------------------
__global__ __launch_bounds__(WAVES_PER_BLOCK * 32)
void sparse_embed_kernel(const float* __restrict__ values,
                         const float* __restrict__ weight,
                         const float* __restrict__ bias,
                         const int*   __restrict__ row_ids,
                         const int*   __restrict__ col_ids,
                         const int*   __restrict__ row_start,   // may be null -> fallback
                         float*       __restrict__ out,
                         int nnz, int batch)
{
    __shared__ float lds[WAVES_PER_BLOCK][2][CHUNK][OUT_DIM];   // 32KB/block

    const int lane = threadIdx.x & 31;
    const int wave = threadIdx.x >> 5;
    const int row  = blockIdx.x * WAVES_PER_BLOCK + wave;
    if (row >= batch) return;

    int segBegin, segEnd;
    if (row_start) {
        segBegin = row_start[row];
        segEnd   = row_start[row + 1];
    } else {
        // Fallback: uniform binary searches (only if workspace was too small)
        int lo = 0, hi = nnz;
        while (lo < hi) { int mid = (lo + hi) >> 1; if (row_ids[mid] < row) lo = mid + 1; else hi = mid; }
        segBegin = lo;
        hi = nnz;
        while (lo < hi) { int mid = (lo + hi) >> 1; if (row_ids[mid] < row + 1) lo = mid + 1; else hi = mid; }
        segEnd = lo;
    }

    float4 acc = make_float4(0.f, 0.f, 0.f, 0.f);
    const int len = segEnd - segBegin;

    if (len > 0) {
        const int nchunks = (len + CHUNK - 1) / CHUNK;

        // Issue CHUNK async 512B row-gathers into buffer `buf`; tail entries are
        // clamped to a valid index (their LDS slots are never read).
        auto issue = [&](int buf, int start) {
            #pragma unroll
            for (int j = 0; j < CHUNK; ++j) {
                int k = start + j;
                if (k > segEnd - 1) k = segEnd - 1;
                const int col = col_ids[k];
                unsigned long long gaddr =
                    (unsigned long long)(weight + (size_t)col * OUT_DIM + lane * 4);
                unsigned ldsOff = (unsigned)(size_t)(&lds[wave][buf][j][lane * 4]);
                asm volatile("global_load_async_to_lds_b128 %0, %1, off"
                             :: "v"(ldsOff), "v"(gaddr) : "memory");
            }
        };

        issue(0, segBegin);
        for (int c = 0; c < nchunks; ++c) {
            if (c + 1 < nchunks) {
                issue((c + 1) & 1, segBegin + (c + 1) * CHUNK);
                // async loads complete in order: <=CHUNK outstanding => chunk c landed
                asm volatile("s_wait_asynccnt %0" :: "i"(CHUNK) : "memory");
            } else {
                asm volatile("s_wait_asynccnt 0" ::: "memory");
            }

            const int base = segBegin + c * CHUNK;
            int m = segEnd - base; if (m > CHUNK) m = CHUNK;
            const int buf = c & 1;

            if (m == CHUNK) {
                // Fast path: cluster the 8 value loads (one wait), then FMA burst.
                float vbuf[CHUNK];
                #pragma unroll
                for (int j = 0; j < CHUNK; ++j) vbuf[j] = values[base + j];
                #pragma unroll
                for (int j = 0; j < CHUNK; ++j) {
                    const float4 w = *(const float4*)(&lds[wave][buf][j][lane * 4]);
                    acc.x = fmaf(w.x, vbuf[j], acc.x);
                    acc.y = fmaf(w.y, vbuf[j], acc.y);
                    acc.z = fmaf(w.z, vbuf[j], acc.z);
                    acc.w = fmaf(w.w, vbuf[j], acc.w);
                }
            } else {
                for (int j = 0; j < m; ++j) {
                    const float v = values[base + j];
                    const float4 w = *(const float4*)(&lds[wave][buf][j][lane * 4]);
                    acc.x = fmaf(w.x, v, acc.x);
                    acc.y = fmaf(w.y, v, acc.y);
                    acc.z = fmaf(w.z, v, acc.z);
                    acc.w = fmaf(w.w, v, acc.w);
                }
            }
        }
    }

    const float4 b = *(const float4*)(bias + lane * 4);
    float4 r;
    r.x = fmaxf(acc.x + b.x, 0.f);
    r.y = fmaxf(acc.y + b.y, 0.f);
    r.z = fmaxf(acc.z + b.z, 0.f);
    r.w = fmaxf(acc.w + b.w, 0.f);
    *(float4*)(out + (size_t)row * OUT_DIM + lane * 4) = r;          // 512B coalesced store
}

extern "C" void kernel_launch(void* const* d_in, const int* in_sizes, int n_in,
                              void* d_out, int out_size, void* d_ws, size_t ws_size,
                              hipStream_t stream) {
    (void)n_in;
    const float* values  = (const float*)d_in[0];
    const float* weight  = (const float*)d_in[1];
    const float* bias    = (const float*)d_in[2];
    const int*   row_ids = (const int*)d_in[3];
    const int*   col_ids = (const int*)d_in[4];
    float* out = (float*)d_out;

    const int nnz   = in_sizes[0];
    const int batch = out_size / OUT_DIM;

    int* row_start = nullptr;
    if (d_ws && ws_size >= (size_t)(batch + 1) * sizeof(int)) {
        row_start = (int*)d_ws;
        const int threads = 256;
        const int grid = (nnz + threads - 1) / threads;
        build_offsets_kernel<<<grid, threads, 0, stream>>>(row_ids, row_start, nnz, batch);
    }

    const int threads = WAVES_PER_BLOCK * 32;
    const int grid = (batch + WAVES_PER_BLOCK - 1) / WAVES_PER_BLOCK;
    sparse_embed_kernel<<<grid, threads, 0, stream>>>(values, weight, bias,
                                                      row_ids, col_ids, row_start, out,
                                                      nnz, batch);
}